// GRACE_60378650247357
// MI455X (gfx1250) — compile-verified
//
#include <hip/hip_runtime.h>
#include <hip/hip_bf16.h>

#define D 128          // feature dim (in = hid = out = 128)
#define LDS_W_FLOATS (128 * 128)   // 64 KB swizzled W tile

typedef __attribute__((ext_vector_type(2))) float v2f;
typedef __attribute__((ext_vector_type(8))) float v8f;

// ---------------- elementwise helpers ----------------

__global__ void __launch_bounds__(256) k_fill(float* __restrict__ p, int n, float v) {
    int i = blockIdx.x * 256 + threadIdx.x;
    if (i < n) p[i] = v;
}

__global__ void __launch_bounds__(256) k_deg_accum(const long long* __restrict__ dst,
                                                   float* __restrict__ deg, int n_edges) {
    int e = blockIdx.x * 256 + threadIdx.x;
    if (e < n_edges) atomicAdd(&deg[(int)dst[e]], 1.0f);
}

__global__ void __launch_bounds__(256) k_rsqrt(float* __restrict__ d, int n) {
    int i = blockIdx.x * 256 + threadIdx.x;
    if (i < n) d[i] = rsqrtf(d[i]);
}

// ---------------- GEMM: H = X @ W via V_WMMA_F32_16X16X4_F32 ----------------
// One wave computes a 16x128 output tile (8 accumulators of v8f).
// W (128x128) staged in LDS, transposed + rotated:
//   sW[n*128 + ((k + 4n) & 127)] = W[k*128 + n]
// -> B-operand reads are LDS-bank-conflict-free and 8B aligned.

__global__ void __launch_bounds__(256) k_gemm_wmma(const float* __restrict__ X,
                                                   const float* __restrict__ W,
                                                   float* __restrict__ H,
                                                   int n_rows) {
    __shared__ float sW[LDS_W_FLOATS];
    for (int i = threadIdx.x; i < 128 * 128; i += 256) {
        int k = i >> 7;
        int n = i & 127;
        sW[n * 128 + ((k + 4 * n) & 127)] = W[i];
    }
    __syncthreads();

    const int wave  = threadIdx.x >> 5;
    const int lane  = threadIdx.x & 31;
    const int tile  = blockIdx.x * 8 + wave;      // 16-row tile index
    const int nTile = n_rows >> 4;                // n_rows divisible by 16
    if (tile >= nTile) return;

    const int m  = lane & 15;   // row within tile (A), col within N-tile (B/C/D)
    const int kh = lane >> 4;   // K-half selector

    const float* xrow = X + (size_t)(tile * 16 + m) * D + kh * 2;

    v8f acc[8] = {};   // 8 N-tiles of 16 cols => 16x128 per wave

#pragma unroll 2
    for (int k0 = 0; k0 < D; k0 += 4) {
        // A: 16x4 f32 tile. lane m, VGPRs = K {k0+2*kh, k0+2*kh+1}
        v2f a = *(const v2f*)(xrow + k0);
        const int kk = k0 + kh * 2;
#pragma unroll
        for (int j = 0; j < 8; ++j) {
            const int n = j * 16 + m;
            // B: 4x16 f32 tile. lane n = col, VGPRs = K {kk, kk+1}
            v2f b = *(const v2f*)&sW[n * 128 + ((kk + 4 * n) & 127)];
            acc[j] = __builtin_amdgcn_wmma_f32_16x16x4_f32(
                false, a, false, b, (short)0, acc[j], false, false);
        }
    }

    // C/D layout: VGPR v -> row M = v + 8*kh, col N = lane&15 (per N-tile)
    float* out = H + (size_t)tile * 16 * D;
#pragma unroll
    for (int j = 0; j < 8; ++j) {
#pragma unroll
        for (int v = 0; v < 8; ++v) {
            out[(size_t)(v + kh * 8) * D + j * 16 + m] = acc[j][v];
        }
    }
}

// ---------------- edge message passing: acc[dst] += h[src] * dinv[src]*dinv[dst] ----
// one block (128 threads) per edge; lane = feature -> coalesced gather + coalesced
// global_atomic_add_f32 into the destination row (resolved in L2).

__global__ void __launch_bounds__(128) k_scatter(const float* __restrict__ h,
                                                 float* __restrict__ acc,
                                                 const long long* __restrict__ src,
                                                 const long long* __restrict__ dst,
                                                 const float* __restrict__ dinv,
                                                 int n_edges) {
    int e = blockIdx.x;
    if (e >= n_edges) return;
    int s = (int)src[e];
    int d = (int)dst[e];
    float norm = dinv[s] * dinv[d];
    int f = threadIdx.x;
    float v = h[(size_t)s * D + f] * norm;
    atomicAdd(&acc[(size_t)d * D + f], v);
}

// ---------------- epilogue: self-loop + bias (+ optional ReLU), in place ----------

__global__ void __launch_bounds__(256) k_epilogue(float* __restrict__ acc,
                                                  const float* __restrict__ h,
                                                  const float* __restrict__ dinv,
                                                  const float* __restrict__ bias,
                                                  int n_nodes, int do_relu) {
    int idx = blockIdx.x * 256 + threadIdx.x;
    int i = idx >> 7;
    if (i >= n_nodes) return;
    int f = idx & 127;
    float di = dinv[i];
    float v = acc[idx] + h[idx] * di * di + bias[f];
    acc[idx] = do_relu ? fmaxf(v, 0.0f) : v;
}

// ---------------- driver ----------------

extern "C" void kernel_launch(void* const* d_in, const int* in_sizes, int n_in,
                              void* d_out, int out_size, void* d_ws, size_t ws_size,
                              hipStream_t stream) {
    const float*     x   = (const float*)d_in[0];
    const long long* ei  = (const long long*)d_in[1];  // int64 edge_index [2, E]
    const float*     W1  = (const float*)d_in[2];
    const float*     b1  = (const float*)d_in[3];
    const float*     W2  = (const float*)d_in[4];
    const float*     b2  = (const float*)d_in[5];
    float*           out = (float*)d_out;

    const int N = in_sizes[0] / D;   // 100000
    const int E = in_sizes[1] / 2;   // 600000
    const long long* src = ei;
    const long long* dst = ei + E;

    // workspace: dinv[N] | bufA[N*D] | bufB[N*D]
    char* ws = (char*)d_ws;
    float* dinv = (float*)ws;
    size_t off = (((size_t)N * sizeof(float)) + 511) & ~(size_t)511;
    float* bufA = (float*)(ws + off);
    float* bufB = (float*)(ws + off + (size_t)N * D * sizeof(float));

    const int nElem = N * D;
    const int gN    = (N + 255) / 256;
    const int gE    = (E + 255) / 256;
    const int gElem = (nElem + 255) / 256;
    const int gGemm = ((N / 16) + 7) / 8;

    // symmetric-normalization degrees (self-loop contributes the initial 1.0)
    k_fill     <<<gN, 256, 0, stream>>>(dinv, N, 1.0f);
    k_deg_accum<<<gE, 256, 0, stream>>>(dst, dinv, E);
    k_rsqrt    <<<gN, 256, 0, stream>>>(dinv, N);

    // ---- layer 1: h1 = x @ W1 ; agg ; +b1 ; ReLU ----
    k_gemm_wmma<<<gGemm, 256, 0, stream>>>(x, W1, bufA, N);
    k_fill     <<<gElem, 256, 0, stream>>>(bufB, nElem, 0.0f);
    k_scatter  <<<E, 128, 0, stream>>>(bufA, bufB, src, dst, dinv, E);
    k_epilogue <<<gElem, 256, 0, stream>>>(bufB, bufA, dinv, b1, N, 1);

    // ---- layer 2: h2 = z1 @ W2 ; agg into d_out ; +b2 ----
    k_gemm_wmma<<<gGemm, 256, 0, stream>>>(bufB, W2, bufA, N);
    k_fill     <<<gElem, 256, 0, stream>>>(out, nElem, 0.0f);
    k_scatter  <<<E, 128, 0, stream>>>(bufA, out, src, dst, dinv, E);
    k_epilogue <<<gElem, 256, 0, stream>>>(out, bufA, dinv, b2, N, 0);
}